// AllAtomFAPE_34540126994458
// MI455X (gfx1250) — compile-verified
//
#include <hip/hip_runtime.h>
#include <hip/hip_bf16.h>
#include <math.h>

// ---- problem constants (from reference setup_inputs) ----
#define NFRAMES 3072   // b*N*F = 1*384*8
#define NATOMS  5376   // b*N*A = 1*384*14
#define D_CLAMP 10.0f
#define EPS_    1e-4f
#define ZSCALE  10.0f

#define FTILES  (NFRAMES / 16)    // 192
#define ATILES  (NATOMS  / 16)    // 336
#define ACHUNK  8                 // atom tiles per wave
#define ACHUNKS (ATILES / ACHUNK) // 42
#define NPART   (FTILES * ACHUNKS)

typedef __attribute__((ext_vector_type(2))) float v2f;
typedef __attribute__((ext_vector_type(8))) float v8f;

// One wave per block. blockIdx.x = frame tile (16 frames), blockIdx.y = chunk of 8 atom tiles.
// diff(f,a) = M_f * x_a + bias_f, with M_f (3x6) = [pR^T | -tR^T], x_a = [pp_a ; tp_a].
// GEMM per tile via V_WMMA_F32_16X16X4_F32: 3 component accumulators x 2 K-chunks.
__global__ __launch_bounds__(32)
void fape_wmma_kernel(const float* __restrict__ pR, const float* __restrict__ pt,
                      const float* __restrict__ pp, const float* __restrict__ am,
                      const float* __restrict__ tR, const float* __restrict__ tt,
                      const float* __restrict__ tp, float* __restrict__ partials)
{
    const int lane = threadIdx.x;       // 0..31
    const int l16  = lane & 15;
    const bool hi  = lane >= 16;
    const int ft   = blockIdx.x;

    // ---------- A operands (16 frames x K=8, fp32), constant over the atom loop ----------
    // Layout (ISA 7.12.2, 32-bit A 16x4): lanes 0-15 -> M=lane, v0=K(4q), v1=K(4q+1);
    //                                     lanes16-31 -> M=lane-16, v0=K(4q+2), v1=K(4q+3).
    // K meaning: K=0..2 -> pR[j][c] (j=K), K=3..5 -> -tR[j][c] (j=K-3), K=6,7 -> 0 (pad).
    const int fA = ft * 16 + l16;
    const float* Rp = pR + fA * 9;
    const float* Rt = tR + fA * 9;
    v2f a0[3], a1[3];
#pragma unroll
    for (int c = 0; c < 3; ++c) {
        float k0 = Rp[0 + c], k1 = Rp[3 + c], k2 = Rp[6 + c];
        float k3 = -Rt[0 + c], k4 = -Rt[3 + c], k5 = -Rt[6 + c];
        a0[c].x = hi ? k2 : k0;
        a0[c].y = hi ? k3 : k1;
        a1[c].x = hi ? 0.f : k4;
        a1[c].y = hi ? 0.f : k5;
    }

    // ---------- per-row bias: (pti - tti)_c = (tR^T tt - pR^T pt)_c ----------
    // D layout: VGPR r holds M = r + (hi ? 8 : 0), so each lane needs 8 frames' biases.
    float bias[8][3];
#pragma unroll
    for (int r = 0; r < 8; ++r) {
        const int f = ft * 16 + r + (hi ? 8 : 0);
        const float* Rpf = pR + f * 9;
        const float* Rtf = tR + f * 9;
        const float* ptf = pt + f * 3;
        const float* ttf = tt + f * 3;
#pragma unroll
        for (int c = 0; c < 3; ++c) {
            float cp = Rpf[0 + c] * ptf[0] + Rpf[3 + c] * ptf[1] + Rpf[6 + c] * ptf[2];
            float ct = Rtf[0 + c] * ttf[0] + Rtf[3 + c] * ttf[1] + Rtf[6 + c] * ttf[2];
            bias[r][c] = ct - cp;
        }
    }

    const float dclamp_sq = D_CLAMP * D_CLAMP;   // min(sqrt(s),10) == sqrt(min(s,100))

    // ---------- loop over 8 atom tiles ----------
    float acc = 0.f;
#pragma unroll 1
    for (int t = 0; t < ACHUNK; ++t) {
        const int atile = blockIdx.y * ACHUNK + t;
        const int a = atile * 16 + l16;
        const float* ppv = pp + a * 3;
        const float* tpv = tp + a * 3;
        // B operand (K=8 x 16 atoms): lanes 0-15 -> N=lane, v0=K(4q), v1=K(4q+1);
        //                             lanes16-31 -> N=lane-16, v0=K(4q+2), v1=K(4q+3).
        float p0 = ppv[0], p1 = ppv[1], p2 = ppv[2];
        float q0 = tpv[0], q1 = tpv[1], q2 = tpv[2];
        v2f b0, b1;
        b0.x = hi ? p2 : p0;
        b0.y = hi ? q0 : p1;
        b1.x = hi ? 0.f : q1;
        b1.y = hi ? 0.f : q2;
        const float mask = am[a];

        v8f d0 = {}, d1 = {}, d2 = {};
        d0 = __builtin_amdgcn_wmma_f32_16x16x4_f32(false, a0[0], false, b0, (short)0, d0, false, false);
        d1 = __builtin_amdgcn_wmma_f32_16x16x4_f32(false, a0[1], false, b0, (short)0, d1, false, false);
        d2 = __builtin_amdgcn_wmma_f32_16x16x4_f32(false, a0[2], false, b0, (short)0, d2, false, false);
        d0 = __builtin_amdgcn_wmma_f32_16x16x4_f32(false, a1[0], false, b1, (short)0, d0, false, false);
        d1 = __builtin_amdgcn_wmma_f32_16x16x4_f32(false, a1[1], false, b1, (short)0, d1, false, false);
        d2 = __builtin_amdgcn_wmma_f32_16x16x4_f32(false, a1[2], false, b1, (short)0, d2, false, false);

        // epilogue: per element (M = r + hi*8, N = l16)
        // dsq chains lower to v_pk_add/pk_fma pairs; clamp in squared domain, then a
        // single raw v_sqrt_f32 (arg >= EPS_, so no denormal-rescue sequence needed).
#pragma unroll
        for (int r = 0; r < 8; ++r) {
            float dx = d0[r] + bias[r][0];
            float dy = d1[r] + bias[r][1];
            float dz = d2[r] + bias[r][2];
            float s  = fmaf(dx, dx, fmaf(dy, dy, fmaf(dz, dz, EPS_)));
            s = fminf(s, dclamp_sq);
            float dist = __builtin_amdgcn_sqrtf(s);   // bare v_sqrt_f32
            acc = fmaf(dist, mask, acc);
        }
    }

    // ---------- wave reduction (wave32) ----------
#pragma unroll
    for (int off = 16; off > 0; off >>= 1)
        acc += __shfl_xor(acc, off, 32);
    if (lane == 0)
        partials[blockIdx.y * gridDim.x + blockIdx.x] = acc;
}

// Deterministic fixed-order final reduction + normalization (single block).
__global__ __launch_bounds__(256)
void fape_finalize_kernel(const float* __restrict__ partials,
                          const float* __restrict__ am,
                          float* __restrict__ out)
{
    __shared__ float s1[256];
    __shared__ float s2[256];
    const int tid = threadIdx.x;
    float s = 0.f;
    for (int i = tid; i < NPART; i += 256) s += partials[i];
    float m = 0.f;
    for (int i = tid; i < NATOMS; i += 256) m += am[i];
    s1[tid] = s;
    s2[tid] = m;
    __syncthreads();
    for (int w = 128; w > 0; w >>= 1) {
        if (tid < w) { s1[tid] += s1[tid + w]; s2[tid] += s2[tid + w]; }
        __syncthreads();
    }
    if (tid == 0) {
        float total = s1[0];
        float msum  = s2[0];
        out[0] = total / (ZSCALE * (EPS_ + (float)NFRAMES) * (EPS_ + msum));
    }
}

extern "C" void kernel_launch(void* const* d_in, const int* in_sizes, int n_in,
                              void* d_out, int out_size, void* d_ws, size_t ws_size,
                              hipStream_t stream) {
    const float* pR = (const float*)d_in[0];  // (1,384,8,3,3)
    const float* pt = (const float*)d_in[1];  // (1,384,8,3)
    const float* pp = (const float*)d_in[2];  // (1,384,14,3)
    const float* am = (const float*)d_in[3];  // (1,384,14)
    const float* tR = (const float*)d_in[4];
    const float* tt = (const float*)d_in[5];
    const float* tp = (const float*)d_in[6];
    float* partials = (float*)d_ws;           // NPART floats (rewritten every call)
    float* out = (float*)d_out;

    dim3 grid(FTILES, ACHUNKS);
    fape_wmma_kernel<<<grid, 32, 0, stream>>>(pR, pt, pp, am, tR, tt, tp, partials);
    fape_finalize_kernel<<<1, 256, 0, stream>>>(partials, am, out);
}